// HAN_28802050687806
// MI455X (gfx1250) — compile-verified
//
#include <hip/hip_runtime.h>
#include <math.h>

typedef __attribute__((ext_vector_type(2))) float v2f;
typedef __attribute__((ext_vector_type(8))) float v8f;

#define HID 128
#define INC 64

// ---------------------------------------------------------------- utilities
__global__ void fill_f32(float* __restrict__ p, float v, int n) {
    int t = blockIdx.x * blockDim.x + threadIdx.x;
    if (t < n) p[t] = v;
}

__device__ __forceinline__ float leaky02(float x) { return x >= 0.f ? x : 0.2f * x; }

// monotonic float atomic-max: positive floats ordered as int, negatives as uint (reversed)
__device__ __forceinline__ void atomicMaxF(float* a, float v) {
    if (v >= 0.f) atomicMax((int*)a, __float_as_int(v));
    else          atomicMin((unsigned int*)a, __float_as_uint(v));
}

// ------------------------------------------------- projection: H = X*W + b
// grid.x = ceil(N/16); block = 256 (8 waves, one 16x16 col-tile each)
__global__ __launch_bounds__(256) void proj_wmma(const float* __restrict__ X,
                                                 const float* __restrict__ W,
                                                 const float* __restrict__ bias,
                                                 float* __restrict__ H, int N) {
    __shared__ float sW[INC * HID];                       // 32 KB
    for (int i = threadIdx.x; i < INC * HID; i += 256) sW[i] = W[i];
    __syncthreads();

    const int wave = threadIdx.x >> 5;                    // column tile 0..7
    const int lane = threadIdx.x & 31;
    const int half = lane >> 4;                           // 0 / 1
    const int l16  = lane & 15;
    const int rowTile = blockIdx.x;
    int row  = rowTile * 16 + l16;
    int rowc = row < N ? row : N - 1;                     // clamped loads
    const int colBase = wave * 16;
    const float* xrow = X + (size_t)rowc * INC;

    v8f acc = {};
#pragma unroll
    for (int kb = 0; kb < INC; kb += 4) {
        const int k = kb + 2 * half;
        v2f a;  a.x = xrow[k];                a.y = xrow[k + 1];
        v2f bm; bm.x = sW[k * HID + colBase + l16];
                bm.y = sW[(k + 1) * HID + colBase + l16];
        acc = __builtin_amdgcn_wmma_f32_16x16x4_f32(false, a, false, bm,
                                                    (short)0, acc, false, false);
    }
    const int col = colBase + l16;
    const float bv = bias[col];
#pragma unroll
    for (int i = 0; i < 8; ++i) {
        int r = rowTile * 16 + i + 8 * half;              // C/D layout (ISA 7.12.2)
        if (r < N) H[(size_t)r * HID + col] = acc[i] + bv;
    }
}

// ---------------------------------- per-node attention logits a[n,h]=<h_nh,att_h>
__global__ void node_att(const float* __restrict__ H, const float* __restrict__ att,
                         float* __restrict__ out, int N) {
    int t = blockIdx.x * blockDim.x + threadIdx.x;
    if (t >= N * 8) return;
    int n = t >> 3, h = t & 7;
    const float* hp = H + (size_t)n * HID + h * 16;
    const float* ap = att + h * 16;
    float s = 0.f;
#pragma unroll
    for (int d = 0; d < 16; ++d) s += hp[d] * ap[d];
    out[t] = s;
}

// ------------------------------------------------- segment softmax, pass 1: max
__global__ void edge_max(const int* __restrict__ src, const int* __restrict__ dst,
                         const float* __restrict__ as_, const float* __restrict__ ad_,
                         float* __restrict__ m, int E) {
    int t = blockIdx.x * blockDim.x + threadIdx.x;
    if (t >= E * 8) return;
    int e = t >> 3, h = t & 7;
    float v = leaky02(as_[src[e] * 8 + h] + ad_[dst[e] * 8 + h]);
    atomicMaxF(&m[dst[e] * 8 + h], v);
}

// ------------------------------------------------- pass 2: sum of exp
__global__ void edge_sum(const int* __restrict__ src, const int* __restrict__ dst,
                         const float* __restrict__ as_, const float* __restrict__ ad_,
                         const float* __restrict__ m, float* __restrict__ s, int E) {
    int t = blockIdx.x * blockDim.x + threadIdx.x;
    if (t >= E * 8) return;
    int e = t >> 3, h = t & 7;
    int d = dst[e];
    float v = leaky02(as_[src[e] * 8 + h] + ad_[d * 8 + h]);
    atomicAdd(&s[d * 8 + h], expf(v - m[d * 8 + h]));
}

// ------------------------------------------------- pass 3: weighted scatter-add
__global__ void edge_agg(const int* __restrict__ src, const int* __restrict__ dst,
                         const float* __restrict__ as_, const float* __restrict__ ad_,
                         const float* __restrict__ m, const float* __restrict__ ssum,
                         const float* __restrict__ Hsrc, float* __restrict__ agg, int E) {
    int t = blockIdx.x * blockDim.x + threadIdx.x;
    if (t >= E * HID) return;
    int e = t >> 7, c = t & 127, h = c >> 4;
    int sn = src[e], dn = dst[e];
    float v = leaky02(as_[sn * 8 + h] + ad_[dn * 8 + h]);
    float w = expf(v - m[dn * 8 + h]) / (ssum[dn * 8 + h] + 1e-16f);
    atomicAdd(&agg[(size_t)dn * HID + c], Hsrc[(size_t)sn * HID + c] * w);
}

// --------------------------------- relu in place + per-column sum (for pooled)
__global__ void relu_colsum(float* __restrict__ agg, float* __restrict__ colsum, int N) {
    int c  = threadIdx.x;                                 // 128 threads
    int r0 = blockIdx.x * 256;
    int r1 = r0 + 256 < N ? r0 + 256 : N;
    float acc = 0.f;
    for (int r = r0; r < r1; ++r) {
        float v = agg[(size_t)r * HID + c];
        v = v > 0.f ? v : 0.f;
        agg[(size_t)r * HID + c] = v;
        acc += v;
    }
    atomicAdd(&colsum[c], acc);
}

// -------- semantic GEMM: tsum[c] += sum_rows tanh(stk @ k_w + k_b)   (WMMA)
// grid.x = ceil(N/16); block = 256 (8 waves = 8 col tiles)
__global__ __launch_bounds__(256) void tanh_gemm_wmma(const float* __restrict__ stk,
                                                      const float* __restrict__ kw,
                                                      const float* __restrict__ kb,
                                                      float* __restrict__ tsum, int N) {
    __shared__ float sKW[HID * HID];                      // 64 KB
    for (int i = threadIdx.x; i < HID * HID; i += 256) sKW[i] = kw[i];
    __syncthreads();

    const int wave = threadIdx.x >> 5;
    const int lane = threadIdx.x & 31;
    const int half = lane >> 4;
    const int l16  = lane & 15;
    const int rowTile = blockIdx.x;
    int row  = rowTile * 16 + l16;
    int rowc = row < N ? row : N - 1;
    const int colBase = wave * 16;
    const float* srow = stk + (size_t)rowc * HID;

    v8f acc = {};
#pragma unroll
    for (int kb2 = 0; kb2 < HID; kb2 += 4) {
        const int k = kb2 + 2 * half;
        v2f a;  a.x = srow[k];               a.y = srow[k + 1];
        v2f bm; bm.x = sKW[k * HID + colBase + l16];
                bm.y = sKW[(k + 1) * HID + colBase + l16];
        acc = __builtin_amdgcn_wmma_f32_16x16x4_f32(false, a, false, bm,
                                                    (short)0, acc, false, false);
    }
    const int col = colBase + l16;
    const float kbv = kb[col];
    float part = 0.f;
#pragma unroll
    for (int i = 0; i < 8; ++i) {
        int r = rowTile * 16 + i + 8 * half;
        if (r < N) part += tanhf(acc[i] + kbv);
    }
    atomicAdd(&tsum[col], part);
}

// ------------------------------------------------------------------ finalize
__global__ void finalize(const float* __restrict__ tsum, const float* __restrict__ colsum,
                         const float* __restrict__ q, const float* __restrict__ lw,
                         const float* __restrict__ lb, float* __restrict__ out,
                         int N0, int N1) {
    __shared__ float red[HID];
    __shared__ float score[4], attn[4];
    __shared__ float pooled[HID];
    int c = threadIdx.x;                                  // 128 threads
    float invN[4] = {1.f / N0, 1.f / N1, 1.f / N0, 1.f / N1};
    for (int mp = 0; mp < 4; ++mp) {
        red[c] = q[c] * tsum[mp * HID + c];
        __syncthreads();
        for (int off = 64; off > 0; off >>= 1) {
            if (c < off) red[c] += red[c + off];
            __syncthreads();
        }
        if (c == 0) score[mp] = red[0] * invN[mp];
        __syncthreads();
    }
    if (c == 0) {
        // metapath order: outs[cell0]=[u00(0), b10(2)], outs[cell1]=[u11(1), b21(3)]
        float m0 = fmaxf(score[0], score[2]);
        float e0 = expf(score[0] - m0), e2 = expf(score[2] - m0);
        attn[0] = e0 / (e0 + e2); attn[2] = e2 / (e0 + e2);
        float m1 = fmaxf(score[1], score[3]);
        float e1 = expf(score[1] - m1), e3 = expf(score[3] - m1);
        attn[1] = e1 / (e1 + e3); attn[3] = e3 / (e1 + e3);
    }
    __syncthreads();
    pooled[c] = attn[0] * colsum[0 * HID + c] + attn[2] * colsum[2 * HID + c]
              + attn[1] * colsum[1 * HID + c] + attn[3] * colsum[3 * HID + c];
    __syncthreads();
    if (c < 2) {
        float s = lb[c];
        for (int k = 0; k < HID; ++k) s += pooled[k] * lw[k * 2 + c];
        out[c] = 1.f / (1.f + expf(-s));
    }
}

// ------------------------------------------------------------------ launcher
static inline int gblk(long long n, int b) { return (int)((n + b - 1) / b); }

extern "C" void kernel_launch(void* const* d_in, const int* in_sizes, int n_in,
                              void* d_out, int out_size, void* d_ws, size_t ws_size,
                              hipStream_t stream) {
    const float* x[3] = {(const float*)d_in[0], (const float*)d_in[1], (const float*)d_in[2]};
    int N[3] = {in_sizes[0] / INC, in_sizes[1] / INC, in_sizes[2] / INC};
    const int* ei[4]; int E[4];
    for (int i = 0; i < 4; ++i) { ei[i] = (const int*)d_in[3 + i]; E[i] = in_sizes[3 + i] / 2; }
    const float* W[3]  = {(const float*)d_in[7], (const float*)d_in[9],  (const float*)d_in[11]};
    const float* bb[3] = {(const float*)d_in[8], (const float*)d_in[10], (const float*)d_in[12]};
    const float* as_[4] = {(const float*)d_in[13], (const float*)d_in[15],
                           (const float*)d_in[17], (const float*)d_in[19]};
    const float* ad_[4] = {(const float*)d_in[14], (const float*)d_in[16],
                           (const float*)d_in[18], (const float*)d_in[20]};
    const float* kw = (const float*)d_in[21];
    const float* kb = (const float*)d_in[22];
    const float* q  = (const float*)d_in[23];
    const float* lw = (const float*)d_in[24];
    const float* lb = (const float*)d_in[25];

    const int srcT[4] = {0, 1, 1, 2}, dstT[4] = {0, 1, 0, 1};

    // ---- workspace carve (256B aligned)
    char* p = (char*)d_ws;
    auto carve = [&](size_t bytes) -> float* {
        float* r = (float*)p;
        p += (bytes + 255) & ~(size_t)255;
        return r;
    };
    float* h[3];
    for (int i = 0; i < 3; ++i) h[i] = carve((size_t)N[i] * HID * 4);
    int maxN = N[0]; if (N[1] > maxN) maxN = N[1]; if (N[2] > maxN) maxN = N[2];
    float* asb = carve((size_t)maxN * 8 * 4);
    float* adb = carve((size_t)maxN * 8 * 4);
    float* mb  = carve((size_t)maxN * 8 * 4);
    float* sb  = carve((size_t)maxN * 8 * 4);
    float* agg = carve((size_t)maxN * HID * 4);
    float* tsum   = carve(4 * HID * 4);
    float* colsum = carve(4 * HID * 4);

    fill_f32<<<gblk(4 * HID, 256), 256, 0, stream>>>(tsum, 0.f, 4 * HID);
    fill_f32<<<gblk(4 * HID, 256), 256, 0, stream>>>(colsum, 0.f, 4 * HID);

    // ---- node-type projections (WMMA f32 16x16x4)
    for (int i = 0; i < 3; ++i)
        proj_wmma<<<gblk(N[i], 16), 256, 0, stream>>>(x[i], W[i], bb[i], h[i], N[i]);

    // ---- per edge type, sequentially (buffers reused)
    for (int mp = 0; mp < 4; ++mp) {
        int Ns = N[srcT[mp]], Nd = N[dstT[mp]];
        const float* Hs = h[srcT[mp]];
        const float* Hd = h[dstT[mp]];
        const int* srcp = ei[mp];
        const int* dstp = ei[mp] + E[mp];

        node_att<<<gblk((long long)Ns * 8, 256), 256, 0, stream>>>(Hs, as_[mp], asb, Ns);
        node_att<<<gblk((long long)Nd * 8, 256), 256, 0, stream>>>(Hd, ad_[mp], adb, Nd);

        fill_f32<<<gblk((long long)Nd * 8, 256), 256, 0, stream>>>(mb, -__builtin_inff(), Nd * 8);
        fill_f32<<<gblk((long long)Nd * 8, 256), 256, 0, stream>>>(sb, 0.f, Nd * 8);
        fill_f32<<<gblk((long long)Nd * HID, 256), 256, 0, stream>>>(agg, 0.f, Nd * HID);

        edge_max<<<gblk((long long)E[mp] * 8, 256), 256, 0, stream>>>(srcp, dstp, asb, adb, mb, E[mp]);
        edge_sum<<<gblk((long long)E[mp] * 8, 256), 256, 0, stream>>>(srcp, dstp, asb, adb, mb, sb, E[mp]);
        edge_agg<<<gblk((long long)E[mp] * HID, 256), 256, 0, stream>>>(srcp, dstp, asb, adb, mb, sb,
                                                                        Hs, agg, E[mp]);

        relu_colsum<<<gblk(Nd, 256), HID, 0, stream>>>(agg, colsum + mp * HID, Nd);
        tanh_gemm_wmma<<<gblk(Nd, 16), 256, 0, stream>>>(agg, kw, kb, tsum + mp * HID, Nd);
    }

    finalize<<<1, HID, 0, stream>>>(tsum, colsum, q, lw, lb, (float*)d_out, N[0], N[1]);
}